// RecalcDistances_36094905155934
// MI455X (gfx1250) — compile-verified
//
#include <hip/hip_runtime.h>
#include <stdint.h>

typedef float v4f __attribute__((ext_vector_type(4)));
typedef int   v4i __attribute__((ext_vector_type(4)));

#define GLOBAL_AS __attribute__((address_space(1)))
#define LDS_AS    __attribute__((address_space(3)))

#ifndef __has_builtin
#define __has_builtin(x) 0
#endif

#if __has_builtin(__builtin_amdgcn_global_load_async_to_lds_b128)
#define USE_ASYNC_LDS 1
#else
#define USE_ASYNC_LDS 0
#endif

__device__ __forceinline__ void wait_asynccnt0() {
#if __has_builtin(__builtin_amdgcn_s_wait_asynccnt)
  __builtin_amdgcn_s_wait_asynccnt(0);
#else
  asm volatile("s_wait_asynccnt 0" ::: "memory");
#endif
}

// Vectorized path: one thread handles 4 consecutive k of the same vertex.
// Requires (K/4) to be a power of two (true for reference K=64 -> shift 4).
// q = quad index in [0, V*K/4); 32-bit indexing (nq <= 2^31 guaranteed by host).
__global__ __launch_bounds__(256) void
RecalcDist_v4(const float* __restrict__ coords,
              const int* __restrict__ nidx,
              float* __restrict__ out,
              int nq, int qshift)
{
    __shared__ v4i s_nidx[256];
    const int  tid    = threadIdx.x;
    const int  q      = (int)(blockIdx.x * 256u + tid);
    const bool active = q < nq;

    const v4i* __restrict__ nidx4 = (const v4i*)nidx;

#if USE_ASYNC_LDS
    if (active) {
        // GLOBAL_LOAD_ASYNC_TO_LDS_B128: per-lane 16B global -> LDS (ASYNCcnt).
        // LDS address = low 32 bits of the generic pointer (wave-relative
        // offset per the LDS aperture mapping).
        __builtin_amdgcn_global_load_async_to_lds_b128(
            (GLOBAL_AS v4i*)(uintptr_t)(nidx4 + q),
            (LDS_AS    v4i*)(uint32_t)(uintptr_t)(&s_nidx[tid]),
            /*offset=*/0, /*cpol=*/0);
    }
    wait_asynccnt0();   // wave-local: each wave reads only its own staged quads
#else
    if (active) s_nidx[tid] = __builtin_nontemporal_load(nidx4 + q);
#endif

    if (!active) return;

    const v4i idx = s_nidx[tid];
    const int v   = q >> qshift;            // vertex of this quad (qpv pow2)

    const v4f* __restrict__ coords4 = (const v4f*)coords;
    const v4f c = coords4[v];   // L2-resident broadcast (16 threads share v)

    v4f dist;
#pragma unroll
    for (int j = 0; j < 4; ++j) {
        const int i  = idx[j];
        const int si = (i < 0) ? 0 : i;     // safe gather index
        v4f n = coords4[si];                // random gather: hits 3.2MB-in-L2
        if (i < 0) n = (v4f){0.f, 0.f, 0.f, 0.f};  // invalid -> ncoords = 0
        const v4f t = n - c;                // so dist = ||c||^2 when invalid
        dist[j] = t.x * t.x + t.y * t.y + t.z * t.z + t.w * t.w;
    }

    // dist is a write-once stream: NT store keeps L2 free for coords.
    __builtin_nontemporal_store(dist, (v4f*)out + q);
}

// Generic scalar fallback (any K, any V): one thread per (v,k) element.
__global__ __launch_bounds__(256) void
RecalcDist_scalar(const float* __restrict__ coords,
                  const int* __restrict__ nidx,
                  float* __restrict__ out, long n, int K)
{
    const long t = (long)blockIdx.x * 256 + threadIdx.x;
    if (t >= n) return;
    const int v = (int)(t / K);
    const v4f* __restrict__ coords4 = (const v4f*)coords;
    const v4f c = coords4[v];
    const int i  = __builtin_nontemporal_load(nidx + t);
    const int si = (i < 0) ? 0 : i;
    v4f nb = coords4[si];
    if (i < 0) nb = (v4f){0.f, 0.f, 0.f, 0.f};
    const v4f d = nb - c;
    __builtin_nontemporal_store(d.x * d.x + d.y * d.y + d.z * d.z + d.w * d.w,
                                out + t);
}

extern "C" void kernel_launch(void* const* d_in, const int* in_sizes, int n_in,
                              void* d_out, int out_size, void* d_ws, size_t ws_size,
                              hipStream_t stream) {
    const float* coords = (const float*)d_in[0];   // [V, 4] float32
    const int*   nidx   = (const int*)d_in[1];     // [V, K] int32
    float*       out    = (float*)d_out;           // [V, K] float32

    const int  C = 4;
    const long V = (long)in_sizes[0] / C;
    if (V <= 0) return;
    const long K = (long)in_sizes[1] / V;

    const long qpv  = K >> 2;                        // quads per vertex
    const bool pow2 = (K > 0) && ((K & 3) == 0) && ((qpv & (qpv - 1)) == 0);
    const long nq   = V * qpv;

    if (pow2 && nq <= 0x7FFFFFFF) {
        const int qshift = __builtin_ctz((unsigned)qpv);
        const int blocks = (int)((nq + 255) / 256);
        RecalcDist_v4<<<blocks, 256, 0, stream>>>(coords, nidx, out,
                                                  (int)nq, qshift);
    } else {
        const long n      = V * K;
        const int  blocks = (int)((n + 255) / 256);
        RecalcDist_scalar<<<blocks, 256, 0, stream>>>(coords, nidx, out, n, (int)K);
    }
}